// LabelIntensityFilter_36876589203621
// MI455X (gfx1250) — compile-verified
//
#include <hip/hip_runtime.h>

// LabelIntensityFilter (segment mean + threshold scatter filter), MI455X/gfx1250.
// Memory-bound. Traffic plan vs 192MB L2:
//   labels (64MB) read twice  -> temporal (2nd read hits L2)
//   vals   (64MB) read once   -> nontemporal load
//   out    (64MB) write once  -> nontemporal store
// => ~192MB HBM traffic @ 23.3 TB/s ~ 8.3us floor.
// Pipeline: init ws -> LDS-privatized histogram (ds_add_f32) -> 512-entry LUT
// -> apply pass with async global->LDS LUT broadcast (gfx1250 ASYNCcnt path).

#define LIF_NUM_LABELS 512
#define LIF_MIN_I (-0.003f)
#define LIF_MAX_I (0.003f)

#if defined(__has_builtin)
#  if __has_builtin(__builtin_amdgcn_global_load_async_to_lds_b64)
#    define LIF_HAVE_ASYNC_LDS 1
#  endif
#  if __has_builtin(__builtin_amdgcn_s_wait_asynccnt)
#    define LIF_HAVE_WAIT_ASYNC 1
#  endif
#endif

// Native vector types (clang ext vectors) so nontemporal builtins accept them.
typedef int   lif_i4 __attribute__((ext_vector_type(4)));
typedef float lif_f4 __attribute__((ext_vector_type(4)));
// Async-LDS builtin takes pointers to a 2-int vector (per clang diagnostic).
typedef int   lif_v2i __attribute__((vector_size(2 * sizeof(int))));

__device__ __forceinline__ void lif_wait_asynccnt0() {
#if defined(LIF_HAVE_WAIT_ASYNC)
    __builtin_amdgcn_s_wait_asynccnt(0);
#else
    asm volatile("s_wait_asynccnt 0" ::: "memory");
#endif
}

// ---------------------------------------------------------------------------
// Apply kernel FIRST in the TU so the disasm snippet shows the gfx1250 async
// global->LDS broadcast. Block size must be 256 (8 wave32s).
// out[i] = lut[label[i]]  with the 2KB LUT staged in LDS.
// ---------------------------------------------------------------------------
__global__ void LIF_apply_kernel(const int* __restrict__ labels,
                                 const int* __restrict__ lut,
                                 int* __restrict__ out,
                                 int n, int n4) {
    __shared__ int slut[LIF_NUM_LABELS];

#if defined(LIF_HAVE_ASYNC_LDS)
    {
        // 256 lanes x 8B = 2KB: one async b64 per lane, tracked by ASYNCcnt.
        int t2 = threadIdx.x << 1;  // 0..510
        __builtin_amdgcn_global_load_async_to_lds_b64(
            (lif_v2i*)(lut + t2), (lif_v2i*)(&slut[t2]), 0, 0);
        lif_wait_asynccnt0();
    }
#else
    for (int i = threadIdx.x; i < LIF_NUM_LABELS; i += blockDim.x)
        slut[i] = lut[i];
#endif
    __syncthreads();

    const lif_i4* l4 = (const lif_i4*)labels;
    lif_i4*       o4 = (lif_i4*)out;
    const int stride = gridDim.x * blockDim.x;

    for (int i = blockIdx.x * blockDim.x + threadIdx.x; i < n4; i += stride) {
        __builtin_prefetch((const void*)(l4 + i + stride), 0, 0);
        lif_i4 L = l4[i];                 // temporal: 2nd read of labels, L2 hit
        lif_i4 O;
        O.x = slut[L.x];                  // ds_load_b32 gathers
        O.y = slut[L.y];
        O.z = slut[L.z];
        O.w = slut[L.w];
        __builtin_nontemporal_store(O, o4 + i);   // streamed once: TH=NT
    }

    // Scalar tail.
    int tail = n4 << 2;
    if (blockIdx.x == 0 && (int)threadIdx.x < n - tail) {
        int idx = tail + threadIdx.x;
        out[idx] = slut[labels[idx]];
    }
}

// ---------------------------------------------------------------------------
// Histogram: LDS-privatized, interleaved h[2*l]=sum, h[2*l+1]=cnt (adjacent
// banks per label). Counts kept in float (exact: max count 32768 << 2^24).
// Labels loaded temporal (re-read by apply), values nontemporal.
// ---------------------------------------------------------------------------
__global__ void LIF_hist_kernel(const int* __restrict__ labels,
                                const float* __restrict__ vals,
                                float* __restrict__ ghist,
                                int n, int n4) {
    __shared__ float h[2 * LIF_NUM_LABELS];
    for (int i = threadIdx.x; i < 2 * LIF_NUM_LABELS; i += blockDim.x) h[i] = 0.0f;
    __syncthreads();

    const lif_i4* l4 = (const lif_i4*)labels;
    const lif_f4* v4 = (const lif_f4*)vals;
    const int stride = gridDim.x * blockDim.x;

    for (int i = blockIdx.x * blockDim.x + threadIdx.x; i < n4; i += stride) {
        // Prefetch next tile while this one is consumed (global_prefetch_b8).
        __builtin_prefetch((const void*)(l4 + i + stride), 0, 0);
        __builtin_prefetch((const void*)(v4 + i + stride), 0, 0);

        lif_i4 L = l4[i];                               // global_load_b128
        lif_f4 V = __builtin_nontemporal_load(v4 + i);  // b128, TH=NT

        atomicAdd(&h[2 * L.x],     V.x);
        atomicAdd(&h[2 * L.x + 1], 1.0f);
        atomicAdd(&h[2 * L.y],     V.y);
        atomicAdd(&h[2 * L.y + 1], 1.0f);
        atomicAdd(&h[2 * L.z],     V.z);
        atomicAdd(&h[2 * L.z + 1], 1.0f);
        atomicAdd(&h[2 * L.w],     V.w);
        atomicAdd(&h[2 * L.w + 1], 1.0f);
    }

    // Scalar tail (n not multiple of 4), straight to global.
    int tail = n4 << 2;
    if (blockIdx.x == 0 && (int)threadIdx.x < n - tail) {
        int idx = tail + threadIdx.x;
        int l = labels[idx];
        atomicAdd(&ghist[2 * l],     vals[idx]);
        atomicAdd(&ghist[2 * l + 1], 1.0f);
    }

    __syncthreads();
    for (int i = threadIdx.x; i < 2 * LIF_NUM_LABELS; i += blockDim.x) {
        atomicAdd(&ghist[i], h[i]);   // merge: 1024 atomics/block into L2
    }
}

// ---------------------------------------------------------------------------
// Per-label threshold -> replacement LUT. lut[l] = bad ? 0 : l.
// One block, 512 threads (16 wave32s).
// ---------------------------------------------------------------------------
__global__ void LIF_lut_kernel(const float* __restrict__ ghist,
                               int* __restrict__ lut) {
    int l = threadIdx.x;
    if (l < LIF_NUM_LABELS) {
        float s = ghist[2 * l];
        float c = ghist[2 * l + 1];
        float m = s / fmaxf(c, 1.0f);
        bool bad = ((m < LIF_MIN_I) || (m > LIF_MAX_I)) && (l != 0);
        lut[l] = bad ? 0 : l;
    }
}

// ---------------------------------------------------------------------------
// Zero the global histogram (2*512 floats: interleaved {sum, cnt}).
// ---------------------------------------------------------------------------
__global__ void LIF_init_kernel(float* __restrict__ ghist) {
    int i = blockIdx.x * blockDim.x + threadIdx.x;
    if (i < 2 * LIF_NUM_LABELS) ghist[i] = 0.0f;
}

// ---------------------------------------------------------------------------
// Host launcher.
// ---------------------------------------------------------------------------
extern "C" void kernel_launch(void* const* d_in, const int* in_sizes, int n_in,
                              void* d_out, int out_size, void* d_ws, size_t ws_size,
                              hipStream_t stream) {
    const int*   labels = (const int*)d_in[0];
    const float* vals   = (const float*)d_in[1];
    int*         out    = (int*)d_out;     // reference output dtype is int32

    const int n  = in_sizes[0];
    const int n4 = n >> 2;

    // Workspace layout: [2*512 floats ghist][512 ints lut]
    float* ghist = (float*)d_ws;
    int*   lut   = (int*)((char*)d_ws + 2 * LIF_NUM_LABELS * sizeof(float));

    LIF_init_kernel<<<(2 * LIF_NUM_LABELS + 255) / 256, 256, 0, stream>>>(ghist);

    int histBlocks = 1024;   // 262144 threads: ~8MB loads in flight + prefetch;
                             // merge-atomic traffic 4MB into L2, overlapped.
    LIF_hist_kernel<<<histBlocks, 256, 0, stream>>>(labels, vals, ghist, n, n4);

    LIF_lut_kernel<<<1, LIF_NUM_LABELS, 0, stream>>>(ghist, lut);

    int applyBlocks = 2048;
    LIF_apply_kernel<<<applyBlocks, 256, 0, stream>>>(labels, lut, out, n, n4);
}